// FourScanBranch_46540265620234
// MI455X (gfx1250) — compile-verified
//
#include <hip/hip_runtime.h>
#include <hip/hip_bf16.h>

typedef __attribute__((ext_vector_type(2))) float v2f;
typedef __attribute__((ext_vector_type(8))) float v8f;

#define BATCH 4
#define NCH   6
#define HH    128
#define WWID  128
#define HWSZ  (HH*WWID)        // 16384
#define LTOT  (NCH*HWSZ)       // 98304
#define DIN   8
#define DSTATE 16
#define CHUNK 512
#define NCHK  (LTOT/CHUNK)     // 192
#define TGRP  64               // steps per async-staged group
#define NGRP  (CHUNK/TGRP)     // 8

__device__ __forceinline__ float siluf(float x) { return x / (1.0f + __expf(-x)); }
__device__ __forceinline__ float softplusf(float x) { return (x > 20.0f) ? x : log1pf(__expf(x)); }

// ---- CDNA5 async memory->LDS helpers (ASYNCcnt-tracked DMA, ISA ch.10.7) ----
__device__ __forceinline__ unsigned lds_off(const void* p) {
    // generic address of LDS: addr[31:0] == wave-relative LDS byte offset (ISA 10.2)
    return (unsigned)(size_t)p;
}
__device__ __forceinline__ void async_ld16(unsigned ldsoff, const float* gptr) {
    asm volatile("global_load_async_to_lds_b128 %0, %1, off"
                 :: "v"(ldsoff), "v"((unsigned long long)(size_t)gptr)
                 : "memory");
}
__device__ __forceinline__ void wait_async0() {
    asm volatile("s_wait_asynccnt 0" ::: "memory");
}

// ---------------------------------------------------------------------------
// K1: stable argsort of distance-from-center by brute-force ranking.
// ---------------------------------------------------------------------------
__global__ void k_rank(int* __restrict__ sp) {
    int i = blockIdx.x * blockDim.x + threadIdx.x;
    if (i >= HWSZ) return;
    int yi = i >> 7, xi = i & 127;
    int d2i = (yi - 64) * (yi - 64) + (xi - 64) * (xi - 64);
    int rank = 0;
    for (int j = 0; j < HWSZ; ++j) {
        int yj = j >> 7, xj = j & 127;
        int dj = (yj - 64) * (yj - 64) + (xj - 64) * (xj - 64);
        rank += (dj < d2i) || (dj == d2i && j < i);
    }
    sp[rank] = i;
}

// ---------------------------------------------------------------------------
// K2: build the 4 scan permutations from sp.
// ---------------------------------------------------------------------------
__global__ void k_perms(const int* __restrict__ sp, int* __restrict__ perms) {
    int l = blockIdx.x * blockDim.x + threadIdx.x;
    if (l >= LTOT) return;
    int n = l / HWSZ, r = l - n * HWSZ;
    perms[0 * LTOT + l] = n * HWSZ + sp[r];
    int r2 = l / NCH, n2 = l - r2 * NCH;
    int tid = ((r2 & 1) == 0) ? n2 : (NCH - 1 - n2);
    perms[1 * LTOT + l] = tid * HWSZ + sp[r2];
    int lp = LTOT - 1 - l;
    n = lp / HWSZ; r = lp - n * HWSZ;
    perms[2 * LTOT + l] = n * HWSZ + sp[r];
    r2 = lp / NCH; n2 = lp - r2 * NCH;
    tid = ((r2 & 1) == 0) ? n2 : (NCH - 1 - n2);
    perms[3 * LTOT + l] = tid * HWSZ + sp[r2];
}

// ---------------------------------------------------------------------------
// K3: gather + branch dwconv(3) + SiLU -> u (LDS), then in_proj via
// V_WMMA_F32_16X16X4_F32 (exact K=4, full f32). 8 waves, 16 rows each.
// A 16x4 f32: lanes 0-15 M=lane K={0,1}; lanes 16-31 M=lane-16 K={2,3}.
// B 4x16 f32: lanes 0-15 N=lane K={0,1}; lanes 16-31 N=lane-16 K={2,3}.
// ---------------------------------------------------------------------------
__global__ void k_proj1(const float* __restrict__ x, const int* __restrict__ perms,
                        const float* __restrict__ dwW, const float* __restrict__ dwB,
                        const float* __restrict__ inW,
                        float* __restrict__ xm_pre, float* __restrict__ zs) {
    __shared__ float u_s[128][5];
    int b  = blockIdx.y;
    int l0 = blockIdx.x * 128;
    const float* xb = x + (size_t)b * LTOT;

    for (int e = threadIdx.x; e < 512; e += 256) {
        int ll = e >> 2, k = e & 3;
        int l = l0 + ll;
        float acc = dwB[k];
        for (int j = 0; j < 3; ++j) {
            int lj = l + j - 1;
            if (lj >= 0 && lj < LTOT) acc += xb[perms[k * LTOT + lj]] * dwW[k * 3 + j];
        }
        u_s[ll][k] = siluf(acc);
    }
    __syncthreads();

    int wave = threadIdx.x >> 5;
    int lane = threadIdx.x & 31;
    int lo = lane & 15, hi = lane >> 4;

    v2f a, bfr;
    a.x   = u_s[wave * 16 + lo][hi * 2 + 0];
    a.y   = u_s[wave * 16 + lo][hi * 2 + 1];
    bfr.x = inW[lo * 4 + hi * 2 + 0];      // W[e=lo][d=k]
    bfr.y = inW[lo * 4 + hi * 2 + 1];
    v8f c = {};
    c = __builtin_amdgcn_wmma_f32_16x16x4_f32(false, a, false, bfr, (short)0, c, false, false);

    for (int v = 0; v < 8; ++v) {
        int m = v + 8 * hi;
        int l = l0 + wave * 16 + m;
        size_t base = (size_t)b * LTOT + l;
        float val = c[v];
        if (lo < 8) xm_pre[base * 8 + lo] = val;
        else        zs[base * 8 + (lo - 8)] = siluf(val);
    }
}

// ---------------------------------------------------------------------------
// K4: causal depthwise conv(4) along L + SiLU -> xm
// ---------------------------------------------------------------------------
__global__ void k_conv(const float* __restrict__ xm_pre, const float* __restrict__ cW,
                       const float* __restrict__ cB, float* __restrict__ xm) {
    int i = blockIdx.x * blockDim.x + threadIdx.x;
    if (i >= BATCH * LTOT * DIN) return;
    int d = i & 7;
    int l = (i >> 3) % LTOT;
    int b = i / (LTOT * DIN);
    size_t base = (size_t)b * LTOT;
    float acc = cB[d];
    for (int j = 0; j < 4; ++j) {
        int lj = l - 3 + j;
        if (lj >= 0) acc += xm_pre[(base + lj) * 8 + d] * cW[d * 4 + j];
    }
    xm[(base + l) * 8 + d] = siluf(acc);
}

// ---------------------------------------------------------------------------
// K5: x_proj (8 -> 33) as 3 N-tiles, each = two chained f32 16x16x4 WMMAs
// (K blocks 0..3 and 4..7 accumulate through C). Fused rank-1 dt_proj+softplus.
// ---------------------------------------------------------------------------
__global__ void k_proj2(const float* __restrict__ xm, const float* __restrict__ xpW,
                        const float* __restrict__ dtW, const float* __restrict__ dtB,
                        float* __restrict__ dt, float* __restrict__ Bmat,
                        float* __restrict__ Cmat) {
    __shared__ float xd_s[128][33];
    int b  = blockIdx.y;
    int l0 = blockIdx.x * 128;
    int wave = threadIdx.x >> 5;
    int lane = threadIdx.x & 31;
    int lo = lane & 15, hi = lane >> 4;

    size_t rowbase = ((size_t)b * LTOT + l0 + wave * 16 + lo) * 8;
    v2f a0, a1;
    a0.x = xm[rowbase + hi * 2 + 0];
    a0.y = xm[rowbase + hi * 2 + 1];
    a1.x = xm[rowbase + 4 + hi * 2 + 0];
    a1.y = xm[rowbase + 4 + hi * 2 + 1];

    for (int t = 0; t < 3; ++t) {
        int e = t * 16 + lo;
        bool ok = e < 33;
        const float* wrow = xpW + (size_t)(ok ? e : 0) * 8;
        v2f b0, b1;
        b0.x = ok ? wrow[hi * 2 + 0] : 0.0f;
        b0.y = ok ? wrow[hi * 2 + 1] : 0.0f;
        b1.x = ok ? wrow[4 + hi * 2 + 0] : 0.0f;
        b1.y = ok ? wrow[4 + hi * 2 + 1] : 0.0f;
        v8f c = {};
        c = __builtin_amdgcn_wmma_f32_16x16x4_f32(false, a0, false, b0, (short)0, c, false, false);
        c = __builtin_amdgcn_wmma_f32_16x16x4_f32(false, a1, false, b1, (short)0, c, false, false);
        for (int v = 0; v < 8; ++v) {
            int m = wave * 16 + v + 8 * hi;
            if (e < 33) xd_s[m][e] = c[v];
        }
    }
    __syncthreads();

    if (threadIdx.x < 128) {
        size_t base = (size_t)b * LTOT + l0 + threadIdx.x;
        float dtraw = xd_s[threadIdx.x][0];
        for (int d = 0; d < 8; ++d)
            dt[base * 8 + d] = softplusf(dtraw * dtW[d] + dtB[d]);
        for (int s = 0; s < 16; ++s) {
            Bmat[base * 16 + s] = xd_s[threadIdx.x][1 + s];
            Cmat[base * 16 + s] = xd_s[threadIdx.x][17 + s];
        }
    }
}

// ---------------------------------------------------------------------------
// K6: chunked scan phase 1, with double-buffered async-to-LDS staging.
// 128 threads = channel (d,s); 8 groups of 64 steps; DMA of group g+1
// overlaps the 64-step recurrence of group g.
// ---------------------------------------------------------------------------
__global__ void k_scanA(const float* __restrict__ dt, const float* __restrict__ xm,
                        const float* __restrict__ Bmat, const float* __restrict__ Alog,
                        float* __restrict__ aggA, float* __restrict__ aggH) {
    __shared__ __align__(16) float dt_s[2][TGRP * 8];
    __shared__ __align__(16) float xm_s[2][TGRP * 8];
    __shared__ __align__(16) float Bm_s[2][TGRP * 16];
    int chunk = blockIdx.x;
    int b = chunk / NCHK, cc = chunk % NCHK;
    int tid = threadIdx.x;
    int d = tid >> 4, s = tid & 15;
    float Ads = -__expf(Alog[d * 16 + s]);
    size_t base0 = (size_t)b * LTOT + (size_t)cc * CHUNK;

    // prologue: DMA group 0 into buffer 0
    async_ld16(lds_off(&dt_s[0][tid * 4]), dt   + base0 * 8  + tid * 4);
    async_ld16(lds_off(&xm_s[0][tid * 4]), xm   + base0 * 8  + tid * 4);
    async_ld16(lds_off(&Bm_s[0][tid * 4]), Bmat + base0 * 16 + tid * 4);
    async_ld16(lds_off(&Bm_s[0][512 + tid * 4]), Bmat + base0 * 16 + 512 + tid * 4);

    float h = 0.0f, ap = 1.0f;
    for (int g = 0; g < NGRP; ++g) {
        int buf = g & 1;
        wait_async0();
        __syncthreads();
        if (g + 1 < NGRP) {
            int nb = (g + 1) & 1;
            size_t gb = base0 + (size_t)(g + 1) * TGRP;
            async_ld16(lds_off(&dt_s[nb][tid * 4]), dt   + gb * 8  + tid * 4);
            async_ld16(lds_off(&xm_s[nb][tid * 4]), xm   + gb * 8  + tid * 4);
            async_ld16(lds_off(&Bm_s[nb][tid * 4]), Bmat + gb * 16 + tid * 4);
            async_ld16(lds_off(&Bm_s[nb][512 + tid * 4]), Bmat + gb * 16 + 512 + tid * 4);
        }
        for (int t = 0; t < TGRP; ++t) {
            float dtv = dt_s[buf][t * 8 + d];
            float av  = __expf(dtv * Ads);
            float bx  = dtv * Bm_s[buf][t * 16 + s] * xm_s[buf][t * 8 + d];
            h  = av * h + bx;
            ap *= av;
        }
        __syncthreads();
    }
    aggA[(size_t)chunk * 128 + tid] = ap;
    aggH[(size_t)chunk * 128 + tid] = h;
}

// ---------------------------------------------------------------------------
// K7: phase 2: serial exclusive scan over the 192 chunk aggregates.
// ---------------------------------------------------------------------------
__global__ void k_scanB(const float* __restrict__ aggA, const float* __restrict__ aggH,
                        float* __restrict__ hinc) {
    int i = blockIdx.x * blockDim.x + threadIdx.x;
    if (i >= BATCH * 128) return;
    int b = i >> 7, ch = i & 127;
    float h = 0.0f;
    for (int c = 0; c < NCHK; ++c) {
        size_t idx = ((size_t)(b * NCHK + c)) * 128 + ch;
        hinc[idx] = h;
        h = aggA[idx] * h + aggH[idx];
    }
}

// ---------------------------------------------------------------------------
// K8: phase 3 fixup with the same async double-buffering (+Cm stream).
// Emits y[b,l,d] via intra-wave shuffle reduction over s.
// ---------------------------------------------------------------------------
__global__ void k_yfix(const float* __restrict__ dt, const float* __restrict__ xm,
                       const float* __restrict__ Bmat, const float* __restrict__ Cmat,
                       const float* __restrict__ Alog, const float* __restrict__ hinc,
                       float* __restrict__ y) {
    __shared__ __align__(16) float dt_s[2][TGRP * 8];
    __shared__ __align__(16) float xm_s[2][TGRP * 8];
    __shared__ __align__(16) float Bm_s[2][TGRP * 16];
    __shared__ __align__(16) float Cm_s[2][TGRP * 16];
    int chunk = blockIdx.x;
    int b = chunk / NCHK, cc = chunk % NCHK;
    int tid = threadIdx.x;
    int d = tid >> 4, s = tid & 15;
    float Ads = -__expf(Alog[d * 16 + s]);
    float h = hinc[(size_t)chunk * 128 + tid];
    size_t base0 = (size_t)b * LTOT + (size_t)cc * CHUNK;

    async_ld16(lds_off(&dt_s[0][tid * 4]), dt   + base0 * 8  + tid * 4);
    async_ld16(lds_off(&xm_s[0][tid * 4]), xm   + base0 * 8  + tid * 4);
    async_ld16(lds_off(&Bm_s[0][tid * 4]), Bmat + base0 * 16 + tid * 4);
    async_ld16(lds_off(&Bm_s[0][512 + tid * 4]), Bmat + base0 * 16 + 512 + tid * 4);
    async_ld16(lds_off(&Cm_s[0][tid * 4]), Cmat + base0 * 16 + tid * 4);
    async_ld16(lds_off(&Cm_s[0][512 + tid * 4]), Cmat + base0 * 16 + 512 + tid * 4);

    for (int g = 0; g < NGRP; ++g) {
        int buf = g & 1;
        wait_async0();
        __syncthreads();
        if (g + 1 < NGRP) {
            int nb = (g + 1) & 1;
            size_t gb = base0 + (size_t)(g + 1) * TGRP;
            async_ld16(lds_off(&dt_s[nb][tid * 4]), dt   + gb * 8  + tid * 4);
            async_ld16(lds_off(&xm_s[nb][tid * 4]), xm   + gb * 8  + tid * 4);
            async_ld16(lds_off(&Bm_s[nb][tid * 4]), Bmat + gb * 16 + tid * 4);
            async_ld16(lds_off(&Bm_s[nb][512 + tid * 4]), Bmat + gb * 16 + 512 + tid * 4);
            async_ld16(lds_off(&Cm_s[nb][tid * 4]), Cmat + gb * 16 + tid * 4);
            async_ld16(lds_off(&Cm_s[nb][512 + tid * 4]), Cmat + gb * 16 + 512 + tid * 4);
        }
        for (int t = 0; t < TGRP; ++t) {
            size_t l = base0 + (size_t)g * TGRP + t;
            float dtv = dt_s[buf][t * 8 + d];
            float av  = __expf(dtv * Ads);
            float bx  = dtv * Bm_s[buf][t * 16 + s] * xm_s[buf][t * 8 + d];
            h = av * h + bx;
            float prod = h * Cm_s[buf][t * 16 + s];
            prod += __shfl_xor(prod, 8, 32);
            prod += __shfl_xor(prod, 4, 32);
            prod += __shfl_xor(prod, 2, 32);
            prod += __shfl_xor(prod, 1, 32);
            if ((tid & 15) == 0) y[l * 8 + d] = prod;
        }
        __syncthreads();
    }
}

// ---------------------------------------------------------------------------
// K9: gate + out_proj (8->4) + LayerNorm(4).
// ---------------------------------------------------------------------------
__global__ void k_post(const float* __restrict__ y, const float* __restrict__ xm,
                       const float* __restrict__ zs, const float* __restrict__ Dp,
                       const float* __restrict__ oW, const float* __restrict__ lg,
                       const float* __restrict__ lb, float* __restrict__ outb) {
    int i = blockIdx.x * blockDim.x + threadIdx.x;
    if (i >= BATCH * LTOT) return;
    size_t base = (size_t)i;
    float yv[8];
    for (int d = 0; d < 8; ++d)
        yv[d] = (y[base * 8 + d] + xm[base * 8 + d] * Dp[d]) * zs[base * 8 + d];
    float o[4], mu = 0.0f;
    for (int e = 0; e < 4; ++e) {
        float acc = 0.0f;
        for (int d = 0; d < 8; ++d) acc += yv[d] * oW[e * 8 + d];
        o[e] = acc; mu += acc;
    }
    mu *= 0.25f;
    float var = 0.0f;
    for (int e = 0; e < 4; ++e) var += (o[e] - mu) * (o[e] - mu);
    var *= 0.25f;
    float inv = rsqrtf(var + 1e-5f);
    for (int e = 0; e < 4; ++e)
        outb[base * 4 + e] = (o[e] - mu) * inv * lg[e] + lb[e];
}

// ---------------------------------------------------------------------------
// K10: inverse gather + mean over the 4 branches -> final (B,N,H,W).
// ---------------------------------------------------------------------------
__global__ void k_gather(const float* __restrict__ outb, const int* __restrict__ perms,
                         float* __restrict__ outF) {
    int i = blockIdx.x * blockDim.x + threadIdx.x;
    if (i >= BATCH * LTOT) return;
    int b = i / LTOT, l = i - b * LTOT;
    float acc = 0.0f;
    for (int k = 0; k < 4; ++k) {
        int g = perms[k * LTOT + l];
        acc += outb[((size_t)b * LTOT + g) * 4 + k];
    }
    outF[i] = 0.25f * acc;
}

// ---------------------------------------------------------------------------
extern "C" void kernel_launch(void* const* d_in, const int* in_sizes, int n_in,
                              void* d_out, int out_size, void* d_ws, size_t ws_size,
                              hipStream_t stream) {
    (void)in_sizes; (void)n_in; (void)out_size; (void)ws_size;

    const float* x    = (const float*)d_in[0];
    const float* dwW  = (const float*)d_in[1];
    const float* dwB  = (const float*)d_in[2];
    const float* inW  = (const float*)d_in[3];
    const float* cW   = (const float*)d_in[4];
    const float* cB   = (const float*)d_in[5];
    const float* xpW  = (const float*)d_in[6];
    const float* dtW  = (const float*)d_in[7];
    const float* dtB  = (const float*)d_in[8];
    const float* Alog = (const float*)d_in[9];
    const float* Dp   = (const float*)d_in[10];
    const float* oW   = (const float*)d_in[11];
    const float* lg   = (const float*)d_in[12];
    const float* lb   = (const float*)d_in[13];

    char* ws = (char*)d_ws;
    size_t off = 0;
    auto alloc = [&](size_t bytes) -> void* {
        void* p = ws + off;
        off += (bytes + 255) & ~(size_t)255;
        return p;
    };
    int*   sp     = (int*)  alloc((size_t)HWSZ * 4);
    int*   perms  = (int*)  alloc((size_t)4 * LTOT * 4);
    float* xm_pre = (float*)alloc((size_t)BATCH * LTOT * 8 * 4);
    float* xm     = (float*)alloc((size_t)BATCH * LTOT * 8 * 4);
    float* zs     = (float*)alloc((size_t)BATCH * LTOT * 8 * 4);
    float* dt     = (float*)alloc((size_t)BATCH * LTOT * 8 * 4);
    float* Bmat   = (float*)alloc((size_t)BATCH * LTOT * 16 * 4);
    float* Cmat   = (float*)alloc((size_t)BATCH * LTOT * 16 * 4);
    float* aggA   = (float*)alloc((size_t)BATCH * NCHK * 128 * 4);
    float* aggH   = (float*)alloc((size_t)BATCH * NCHK * 128 * 4);
    float* hinc   = (float*)alloc((size_t)BATCH * NCHK * 128 * 4);
    float* y      = (float*)alloc((size_t)BATCH * LTOT * 8 * 4);
    float* outb   = (float*)alloc((size_t)BATCH * LTOT * 4 * 4);

    k_rank  <<<HWSZ / 256, 256, 0, stream>>>(sp);
    k_perms <<<LTOT / 256, 256, 0, stream>>>(sp, perms);
    k_proj1 <<<dim3(LTOT / 128, BATCH), 256, 0, stream>>>(x, perms, dwW, dwB, inW, xm_pre, zs);
    k_conv  <<<(BATCH * LTOT * 8) / 256, 256, 0, stream>>>(xm_pre, cW, cB, xm);
    k_proj2 <<<dim3(LTOT / 128, BATCH), 256, 0, stream>>>(xm, xpW, dtW, dtB, dt, Bmat, Cmat);
    k_scanA <<<BATCH * NCHK, 128, 0, stream>>>(dt, xm, Bmat, Alog, aggA, aggH);
    k_scanB <<<2, 256, 0, stream>>>(aggA, aggH, hinc);
    k_yfix  <<<BATCH * NCHK, 128, 0, stream>>>(dt, xm, Bmat, Cmat, Alog, hinc, y);
    k_post  <<<(BATCH * LTOT) / 256, 256, 0, stream>>>(y, xm, zs, Dp, oW, lg, lb, outb);
    k_gather<<<(BATCH * LTOT) / 256, 256, 0, stream>>>(outb, perms, (float*)d_out);
}